// DsModel_23287312679035
// MI455X (gfx1250) — compile-verified
//
#include <hip/hip_runtime.h>

// Fused add + RMSNorm + (smooth) dynamic int8 quant for MI455X (gfx1250).
// Bandwidth-bound (~1.2 GB traffic per call): CDNA5 async LDS-DMA path
// (global_load_async_to_lds_b128 + s_wait_asynccnt) with software prefetch
// of the next row, b128 loads, and non-temporal b128 streaming stores.

#define HDIM    4096
#define THREADS 256
#define F4R     (HDIM / 4)        // 1024 float4 per row
#define CH      (F4R / THREADS)   // 4 float4 chunks per thread
#define NWAVES  (THREADS / 32)    // 8 waves (wave32)
#define EPSV    1e-6f
#define MAXGRID 2048

typedef float float4v __attribute__((ext_vector_type(4)));
typedef int   int4v   __attribute__((ext_vector_type(4)));

// ---- CDNA5 async global->LDS copy (ASYNCcnt-tracked LDS-DMA) -------------
__device__ __forceinline__ void async_copy_b128(float4v* lds_dst,
                                                const float4v* gsrc) {
#if __has_builtin(__builtin_amdgcn_global_load_async_to_lds_b128)
  __builtin_amdgcn_global_load_async_to_lds_b128(
      (__attribute__((address_space(1))) int4v*)gsrc,
      (__attribute__((address_space(3))) int4v*)lds_dst,
      /*offset=*/0, /*cpol=*/0);
#else
  unsigned lds_off =
      (unsigned)(unsigned long long)(__attribute__((address_space(3))) void*)lds_dst;
  asm volatile("global_load_async_to_lds_b128 %0, %1, off"
               :: "v"(lds_off), "v"(gsrc)
               : "memory");
#endif
}

__device__ __forceinline__ void async_wait0() {
#if __has_builtin(__builtin_amdgcn_s_wait_asynccnt)
  __builtin_amdgcn_s_wait_asynccnt(0);
#else
  asm volatile("s_wait_asynccnt 0" ::: "memory");
#endif
}

// Non-temporal streaming store: outputs are write-once, never re-read, so
// keep them out of the 192 MB L2 that the read stream / DMA prefetch uses.
__device__ __forceinline__ void store_nt(float4v* dst, float4v v) {
  __builtin_nontemporal_store(v, dst);
}

// ---- wave32 reductions ---------------------------------------------------
__device__ __forceinline__ float wave_sum(float v) {
#pragma unroll
  for (int o = 16; o > 0; o >>= 1) v += __shfl_xor(v, o, 32);
  return v;
}
__device__ __forceinline__ float wave_max(float v) {
#pragma unroll
  for (int o = 16; o > 0; o >>= 1) v = fmaxf(v, __shfl_xor(v, o, 32));
  return v;
}

__global__ __launch_bounds__(THREADS)
void fused_add_rmsnorm_dynquant(const float* __restrict__ x1,
                                const float* __restrict__ x2,
                                const float* __restrict__ wgt,
                                const float* __restrict__ smo,
                                float* __restrict__ out,
                                int rows) {
  __shared__ float4v ldsA[F4R];       // x1 row tile (16 KB)
  __shared__ float4v ldsB[F4R];       // x2 row tile (16 KB)
  __shared__ float redS[NWAVES];
  __shared__ float redM1[NWAVES];
  __shared__ float redM3[NWAVES];

  const int tid    = threadIdx.x;
  const int lane   = tid & 31;
  const int wid    = tid >> 5;
  const int stride = gridDim.x;

  const float4v* x1v = (const float4v*)x1;
  const float4v* x2v = (const float4v*)x2;
  const float4v* wv  = (const float4v*)wgt;
  const float4v* smv = (const float4v*)smo;
  float4v* out4 = (float4v*)out;

  // output section offsets (in floats), tuple order of the reference
  const size_t Nf = (size_t)rows * HDIM;
  const size_t Rf = (size_t)rows;
  const size_t o_yOut  = 0;                 // q(smooth)        [N]
  const size_t o_xOut  = Nf;                // x1+x2            [N]
  const size_t o_sc1   = 2 * Nf;            // scale1           [R]
  const size_t o_y1    = 2 * Nf + Rf;       // y (fp32)         [N]
  const size_t o_xOut1 = 3 * Nf + Rf;       // x1+x2            [N]
  const size_t o_yOut2 = 4 * Nf + Rf;       // q(smooth)        [N]
  const size_t o_sc2   = 5 * Nf + Rf;       // scale1           [R]
  const size_t o_xOut3 = 5 * Nf + 2 * Rf;   // x1+x2            [N]
  const size_t o_yOut3 = 6 * Nf + 2 * Rf;   // q(no smooth)     [N]
  const size_t o_sc3   = 7 * Nf + 2 * Rf;   // scale3           [R]

  // Row-invariant vectors: load this thread's slice once, keep in VGPRs.
  float4v wr[CH], sr[CH];
#pragma unroll
  for (int i = 0; i < CH; ++i) {
    const int f = i * THREADS + tid;
    wr[i] = wv[f];
    sr[i] = smv[f];
  }

  // Issue the async DMA for the first row owned by this block.
  int r = blockIdx.x;
  if (r < rows) {
    const size_t base = (size_t)r * F4R;
#pragma unroll
    for (int i = 0; i < CH; ++i) {
      const int f = i * THREADS + tid;
      async_copy_b128(&ldsA[f], &x1v[base + f]);
      async_copy_b128(&ldsB[f], &x2v[base + f]);
    }
  }

  for (; r < rows; r += stride) {
    async_wait0();  // this wave's tiles are resident in LDS

    // x = x1 + x2 and sum of squares (each lane reads only slots it issued)
    float4v xv[CH];
    float ssum = 0.f;
#pragma unroll
    for (int i = 0; i < CH; ++i) {
      const int f = i * THREADS + tid;
      const float4v a = ldsA[f];
      const float4v b = ldsB[f];
      const float4v x = a + b;
      xv[i] = x;
      ssum += x.x * x.x + x.y * x.y + x.z * x.z + x.w * x.w;
    }

    // Prefetch next row via LDS-DMA while we reduce/quantize/store this one.
    const int rn = r + stride;
    if (rn < rows) {
      const size_t base = (size_t)rn * F4R;
#pragma unroll
      for (int i = 0; i < CH; ++i) {
        const int f = i * THREADS + tid;
        async_copy_b128(&ldsA[f], &x1v[base + f]);
        async_copy_b128(&ldsB[f], &x2v[base + f]);
      }
    }

    // block reduction: sum of squares -> rstd
    ssum = wave_sum(ssum);
    if (lane == 0) redS[wid] = ssum;
    __syncthreads();
    float tot = 0.f;
#pragma unroll
    for (int j = 0; j < NWAVES; ++j) tot += redS[j];
    const float inv = rsqrtf(tot * (1.f / HDIM) + EPSV);

    // y = x * rstd * w ; row abs-max of y*smooth and of y
    float4v yv[CH];
    float m1 = 0.f, m3 = 0.f;
#pragma unroll
    for (int i = 0; i < CH; ++i) {
      const float4v y = xv[i] * inv * wr[i];
      const float4v s = y * sr[i];
      yv[i] = y;
      m1 = fmaxf(m1, fmaxf(fmaxf(fabsf(s.x), fabsf(s.y)),
                           fmaxf(fabsf(s.z), fabsf(s.w))));
      m3 = fmaxf(m3, fmaxf(fmaxf(fabsf(y.x), fabsf(y.y)),
                           fmaxf(fabsf(y.z), fabsf(y.w))));
    }
    m1 = wave_max(m1);
    m3 = wave_max(m3);
    if (lane == 0) { redM1[wid] = m1; redM3[wid] = m3; }
    __syncthreads();
    float M1 = 0.f, M3 = 0.f;
#pragma unroll
    for (int j = 0; j < NWAVES; ++j) {
      M1 = fmaxf(M1, redM1[j]);
      M3 = fmaxf(M3, redM3[j]);
    }
    __syncthreads();  // red* buffers reused next iteration

    const float q1 = (M1 > 0.f) ? 127.f / M1 : 0.f;  // s/scale = s*127/max
    const float q3 = (M3 > 0.f) ? 127.f / M3 : 0.f;

    // store all 7 full-size outputs as non-temporal b128 streaming stores
    const size_t rowf4 = (size_t)r * F4R;
#pragma unroll
    for (int i = 0; i < CH; ++i) {
      const int f = i * THREADS + tid;
      const size_t fg = rowf4 + (size_t)f;
      const float4v x = xv[i];
      const float4v y = yv[i];
      const float4v s = y * sr[i];
      const float4v qs = { rintf(s.x * q1), rintf(s.y * q1),
                           rintf(s.z * q1), rintf(s.w * q1) };
      const float4v qn = { rintf(y.x * q3), rintf(y.y * q3),
                           rintf(y.z * q3), rintf(y.w * q3) };
      store_nt(&out4[(o_yOut  >> 2) + fg], qs);
      store_nt(&out4[(o_xOut  >> 2) + fg], x);
      store_nt(&out4[(o_y1    >> 2) + fg], y);
      store_nt(&out4[(o_xOut1 >> 2) + fg], x);
      store_nt(&out4[(o_yOut2 >> 2) + fg], qs);
      store_nt(&out4[(o_xOut3 >> 2) + fg], x);
      store_nt(&out4[(o_yOut3 >> 2) + fg], qn);
    }
    if (tid == 0) {
      const float sc1 = M1 * (1.f / 127.f);
      out[o_sc1 + (size_t)r] = sc1;
      out[o_sc2 + (size_t)r] = sc1;
      out[o_sc3 + (size_t)r] = M3 * (1.f / 127.f);
    }
  }
}

extern "C" void kernel_launch(void* const* d_in, const int* in_sizes, int n_in,
                              void* d_out, int out_size, void* d_ws,
                              size_t ws_size, hipStream_t stream) {
  (void)n_in; (void)out_size; (void)d_ws; (void)ws_size;
  const float* x1 = (const float*)d_in[0];
  const float* x2 = (const float*)d_in[1];
  const float* w  = (const float*)d_in[2];
  const float* ss = (const float*)d_in[3];
  float* out = (float*)d_out;

  const int rows = in_sizes[0] / HDIM;  // B*S = 8192 for the reference shapes
  if (rows <= 0) return;
  const int grid = rows < MAXGRID ? rows : MAXGRID;
  fused_add_rmsnorm_dynquant<<<grid, THREADS, 0, stream>>>(x1, x2, w, ss, out,
                                                           rows);
}